// StructuredGraphAttentionLayer_53987738911432
// MI455X (gfx1250) — compile-verified
//
#include <hip/hip_runtime.h>
#include <hip/hip_bf16.h>
#include <math.h>

// ---------------------------------------------------------------------------
// StructuredGraphAttentionLayer for MI455X (gfx1250, wave32, WMMA + async LDS)
// B=1024, N=110, D=300.  fp32 end-to-end; GEMMs use V_WMMA_F32_16X16X4_F32,
// B-panel staging uses GLOBAL_LOAD_ASYNC_TO_LDS_B128 (ASYNCcnt).
// ---------------------------------------------------------------------------

#define Bb 1024
#define Nn 110
#define Dd 300
#define BN (Bb * Nn)            // 112640 rows (multiple of 64)

typedef __attribute__((ext_vector_type(2))) float v2f;
typedef __attribute__((ext_vector_type(8))) float v8f;
typedef __attribute__((address_space(3))) float lds_f;

#if defined(__has_builtin)
#if __has_builtin(__builtin_amdgcn_wmma_f32_16x16x4_f32)
#define HAVE_WMMA_F32X4 1
#endif
#endif

// D(16x16 f32) += A(16x4 f32) * B(4x16 f32)
__device__ __forceinline__ v8f wmma4(v2f a, v2f b, v8f c) {
#if defined(HAVE_WMMA_F32X4)
  return __builtin_amdgcn_wmma_f32_16x16x4_f32(false, a, false, b, (short)0, c,
                                               false, false);
#else
  c[0] += a[0] * b[0];  // host-pass / missing-builtin placeholder
  return c;
#endif
}

// async 16-byte global -> LDS copy (per-lane), tracked on ASYNCcnt
__device__ __forceinline__ void async_cp_b128(float* lds_dst,
                                              const float* gsrc) {
#if defined(__gfx1250__)
  lds_f* lp = (lds_f*)lds_dst;  // addrspacecast: VGPR holds LDS byte offset
  asm volatile("global_load_async_to_lds_b128 %0, %1, off"
               :
               : "v"(lp), "v"(gsrc)
               : "memory");
#else
  for (int i = 0; i < 4; ++i) lds_dst[i] = gsrc[i];
#endif
}

__device__ __forceinline__ void async_wait() {
#if defined(__gfx1250__)
  asm volatile("s_wait_asynccnt 0x0" ::: "memory");
#endif
}

// ---------------------------------------------------------------------------
// Stage a 16-column B panel, column-major pan[c][k] (k contiguous), from a
// row-major matrix src (column c of the GEMM = row (colBase+c) of src).
// nK = 300 floats per column.  Async fast path when all 16 columns in range.
// ---------------------------------------------------------------------------
__device__ __forceinline__ void stage_panel_cmajor(float* pan,
                                                   const float* src,
                                                   int srcStride, int srcOff,
                                                   int colBase, int tid) {
  if (colBase + 16 <= Dd) {
    for (int q = tid; q < (16 * Dd) / 4; q += 128) {   // 1200 16B chunks
      int c = q / (Dd / 4);
      int k4 = (q - c * (Dd / 4)) * 4;
      async_cp_b128(pan + c * Dd + k4,
                    src + (size_t)(colBase + c) * srcStride + srcOff + k4);
    }
    async_wait();
  } else {
    for (int idx = tid; idx < 16 * Dd; idx += 128) {
      int c = idx / Dd, k = idx - c * Dd;
      int gc = colBase + c;
      pan[idx] = (gc < Dd) ? src[(size_t)gc * srcStride + srcOff + k] : 0.0f;
    }
  }
}

// ---------------------------------------------------------------------------
// K1: h = X(BN x 300) @ W(300 x 300), WMMA-tiled, W panel (k-major) in LDS
// via async-to-LDS.  block = 128 (4 waves), tile 64x16, grid=(BN/64, 19)
// ---------------------------------------------------------------------------
__global__ __launch_bounds__(128) void k_gemm_h(const float* __restrict__ X,
                                                const float* __restrict__ W,
                                                float* __restrict__ Hm) {
  __shared__ float ldsW[Dd * 16];  // [k][c] panel, 19.2 KB
  const int tid = threadIdx.x;
  const int colBase = blockIdx.y * 16;
  if (colBase + 16 <= Dd) {
    for (int q = tid; q < (Dd * 16) / 4; q += 128) {  // 1200 16B chunks
      int k = q >> 2, c4 = (q & 3) * 4;
      async_cp_b128(ldsW + k * 16 + c4, W + (size_t)k * Dd + colBase + c4);
    }
    async_wait();
  } else {
    for (int idx = tid; idx < Dd * 16; idx += 128) {
      int k = idx >> 4, c = idx & 15;
      int gc = colBase + c;
      ldsW[idx] = (gc < Dd) ? W[(size_t)k * Dd + gc] : 0.0f;
    }
  }
  __syncthreads();

  const int wave = tid >> 5, lane = tid & 31;
  const int lrow = lane & 15, lhalf = lane >> 4;
  const int rowBase = blockIdx.x * 64 + wave * 16;
  const float* xr = X + (size_t)(rowBase + lrow) * Dd;

  v8f acc = {};
  for (int kk = 0; kk < Dd / 4; ++kk) {
    v2f a, b;
#pragma unroll
    for (int r = 0; r < 2; ++r) {
      int k = kk * 4 + r + 2 * lhalf;
      a[r] = xr[k];
      b[r] = ldsW[k * 16 + lrow];
    }
    acc = wmma4(a, b, acc);
  }
#pragma unroll
  for (int r = 0; r < 8; ++r) {
    int mm = rowBase + r + 8 * lhalf;
    int c = colBase + lrow;
    if (c < Dd) Hm[(size_t)mm * Dd + c] = acc[r];
  }
}

// ---------------------------------------------------------------------------
// K0b: re_dot[d] = sum_k root_emb[k] * agg_W[d, 300+k]
// ---------------------------------------------------------------------------
__global__ void k_redot(const float* __restrict__ aggW,
                        const float* __restrict__ remb,
                        float* __restrict__ redot) {
  int d = blockIdx.x * blockDim.x + threadIdx.x;
  if (d < Dd) {
    float s = 0.0f;
    for (int k = 0; k < Dd; ++k) s += remb[k] * aggW[(size_t)d * 900 + 300 + k];
    redot[d] = s;
  }
}

// ---------------------------------------------------------------------------
// K2: per-node reductions: root_prob, attn_src, attn_dst. 1 wave / node.
// ---------------------------------------------------------------------------
__global__ __launch_bounds__(128) void k_node(const float* __restrict__ Hm,
                                              const float* __restrict__ rootW,
                                              const float* __restrict__ rootb,
                                              const float* __restrict__ wsrc,
                                              const float* __restrict__ wdst,
                                              float* __restrict__ asrc,
                                              float* __restrict__ adst,
                                              float* __restrict__ rprb) {
  const int node = blockIdx.x * 4 + (threadIdx.x >> 5);
  const int lane = threadIdx.x & 31;
  if (node >= BN) return;
  const float* hr = Hm + (size_t)node * Dd;
  float sr = 0.0f, ss = 0.0f, sd = 0.0f;
  for (int k = lane; k < Dd; k += 32) {
    float hv = hr[k];
    sr += hv * rootW[k];
    float t = tanhf(hv);
    ss += t * wsrc[k];
    sd += t * wdst[k];
  }
#pragma unroll
  for (int m = 16; m >= 1; m >>= 1) {
    sr += __shfl_xor(sr, m, 32);
    ss += __shfl_xor(ss, m, 32);
    sd += __shfl_xor(sd, m, 32);
  }
  if (lane == 0) {
    rprb[node] = sr + rootb[0];
    asrc[node] = ss;
    adst[node] = sd;
  }
}

// ---------------------------------------------------------------------------
// K3: A = exp(clamped leaky attn), R, colsum, LL = diag(colsum+R) - A.
// ---------------------------------------------------------------------------
__global__ __launch_bounds__(128) void k_buildA(const float* __restrict__ adj,
                                                const float* __restrict__ asrc,
                                                const float* __restrict__ adst,
                                                const float* __restrict__ rprb,
                                                float* __restrict__ A,
                                                float* __restrict__ LL,
                                                float* __restrict__ R) {
  const int b = blockIdx.x;
  const int tid = threadIdx.x;
  __shared__ float ssrc[Nn], sm[Nn];
  if (tid < Nn) {
    sm[tid] = 1.0f - adj[((size_t)b * Nn + tid) * Nn];  // mask_ = 1 - adj[b,n,0]
    ssrc[tid] = asrc[b * Nn + tid];
  }
  __syncthreads();
  if (tid < Nn) {
    const int j = tid;
    const float dj = adst[b * Nn + j];
    const float mj = sm[j];
    float cs = 0.0f, ajj = 0.0f;
    for (int i = 0; i < Nn; ++i) {
      float v = ssrc[i] + dj;
      v = (v >= 0.0f) ? v : 0.2f * v;            // leaky relu
      v = v - 50.0f * mj - 50.0f * sm[i];
      v = fmaxf(v, -40.0f);
      float a = expf(v);
      size_t o = ((size_t)b * Nn + i) * Nn + j;
      A[o] = a;
      LL[o] = -a;
      cs += a;
      if (i == j) ajj = a;
    }
    float rj = expf(fmaxf(rprb[b * Nn + j] - 50.0f * mj, -40.0f));
    R[b * Nn + j] = rj;
    LL[((size_t)b * Nn + j) * Nn + j] = cs + rj - ajj;
  }
}

// ---------------------------------------------------------------------------
// K4: in-place Gauss-Jordan inverse of LL (110x110) in LDS (column diag
// dominant => pivot-free).  Writes inverse back, diag d, root_out.
// ---------------------------------------------------------------------------
__global__ __launch_bounds__(256) void k_invert(float* __restrict__ LL,
                                                float* __restrict__ dv,
                                                const float* __restrict__ R,
                                                const float* __restrict__ adj,
                                                float* __restrict__ ro,
                                                float* __restrict__ out_tail) {
  const int b = blockIdx.x;
  const int tid = threadIdx.x;
  __shared__ float s[Nn * (Nn + 1)];  // stride 111 -> odd mod 64 banks
  __shared__ float fc[Nn];
  for (int idx = tid; idx < Nn * Nn; idx += 256) {
    int i = idx / Nn, j = idx - i * Nn;
    s[i * (Nn + 1) + j] = LL[((size_t)b * Nn + i) * Nn + j];
  }
  __syncthreads();

  for (int k = 0; k < Nn; ++k) {
    for (int i = tid; i < Nn; i += 256) fc[i] = s[i * (Nn + 1) + k];
    __syncthreads();
    const float piv = 1.0f / fc[k];
    for (int j = tid; j < Nn; j += 256) {
      float v = (j == k) ? 1.0f : s[k * (Nn + 1) + j];
      s[k * (Nn + 1) + j] = v * piv;
    }
    __syncthreads();
    for (int idx = tid; idx < Nn * Nn; idx += 256) {
      int i = idx / Nn, j = idx - i * Nn;
      if (i != k) {
        float v = (j == k) ? 0.0f : s[i * (Nn + 1) + j];
        s[i * (Nn + 1) + j] = v - fc[i] * s[k * (Nn + 1) + j];
      }
    }
    __syncthreads();
  }

  for (int idx = tid; idx < Nn * Nn; idx += 256) {
    int i = idx / Nn, j = idx - i * Nn;
    LL[((size_t)b * Nn + i) * Nn + j] = s[i * (Nn + 1) + j];  // now LL^{-1}
  }
  for (int n = tid; n < Nn; n += 256) {
    float d = s[n * (Nn + 1) + n];
    dv[b * Nn + n] = d;
    float m = 1.0f - adj[((size_t)b * Nn + n) * Nn];
    float r = (m > 0.0f) ? 0.0f : R[b * Nn + n] * d;
    ro[b * Nn + n] = r;
    out_tail[b * Nn + n] = r;  // second reference output
  }
}

// ---------------------------------------------------------------------------
// K5: attn_out[b,i,j] = mask? 0 : A[b,j,i] * (d[b,i] - LLinv[b,i,j])
// ---------------------------------------------------------------------------
__global__ __launch_bounds__(128) void k_attnout(const float* __restrict__ A,
                                                 const float* __restrict__ LLinv,
                                                 const float* __restrict__ dv,
                                                 const float* __restrict__ adj,
                                                 float* __restrict__ AO) {
  const int b = blockIdx.x, i = blockIdx.y, j = threadIdx.x;
  if (j < Nn) {
    size_t o = ((size_t)b * Nn + i) * Nn + j;
    float mask = 1.0f - adj[o];
    float v = 0.0f;
    if (mask <= 0.0f)
      v = A[((size_t)b * Nn + j) * Nn + i] * (dv[b * Nn + i] - LLinv[o]);
    AO[o] = v;
  }
}

// ---------------------------------------------------------------------------
// K6: dual batched GEMM: feat_c = AO @ X, feat_p = AO^T @ X, shared B frags.
// ---------------------------------------------------------------------------
__global__ __launch_bounds__(128) void k_feat_cp(const float* __restrict__ AO,
                                                 const float* __restrict__ X,
                                                 float* __restrict__ FC,
                                                 float* __restrict__ FP) {
  const int b = blockIdx.y;
  const int wave = threadIdx.x >> 5, lane = threadIdx.x & 31;
  const int t = blockIdx.x * 4 + wave;
  if (t >= 7 * 19) return;  // wave-uniform
  const int mt = t / 19, nt = t - mt * 19;
  const int lrow = lane & 15, lhalf = lane >> 4;
  const float* Ab = AO + (size_t)b * Nn * Nn;
  const float* Xb = X + (size_t)b * Nn * Dd;
  const int row = mt * 16 + lrow;
  const int col = nt * 16 + lrow;

  v8f accC = {}, accP = {};
  for (int kk = 0; kk < 28; ++kk) {
    v2f a, at, bf;
#pragma unroll
    for (int r = 0; r < 2; ++r) {
      int k = kk * 4 + r + 2 * lhalf;
      bool kok = (k < Nn);
      a[r]  = (kok && row < Nn) ? Ab[row * Nn + k] : 0.0f;
      at[r] = (kok && row < Nn) ? Ab[k * Nn + row] : 0.0f;
      bf[r] = (kok && col < Dd) ? Xb[k * Dd + col] : 0.0f;
    }
    accC = wmma4(a, bf, accC);
    accP = wmma4(at, bf, accP);
  }
#pragma unroll
  for (int r = 0; r < 8; ++r) {
    int mm = mt * 16 + r + 8 * lhalf;
    int cc = nt * 16 + lrow;
    if (mm < Nn && cc < Dd) {
      size_t o = ((size_t)b * Nn + mm) * Dd + cc;
      FC[o] = accC[r];
      FP[o] = accP[r];
    }
  }
}

// ---------------------------------------------------------------------------
// K7: fused output GEMM (K=900 over [FC|FP|X] vs agg_W^T) + gate GEMM
// sharing X A-frags.  4 waves/block share one col tile; B panels staged
// column-major in LDS via async-to-LDS (19.2 KB x2).
// grid = (BN/64, 19), block 128.
// ---------------------------------------------------------------------------
__global__ __launch_bounds__(128) void k_final(const float* __restrict__ FC,
                                               const float* __restrict__ FP,
                                               const float* __restrict__ X,
                                               const float* __restrict__ aggW,
                                               const float* __restrict__ aggb,
                                               const float* __restrict__ HW,
                                               const float* __restrict__ Hb,
                                               const float* __restrict__ ro,
                                               const float* __restrict__ redot,
                                               float* __restrict__ out) {
  __shared__ float pB1[16 * Dd];  // agg panel, c-major [c][k]
  __shared__ float pB2[16 * Dd];  // H_W panel,  c-major [c][k]
  const int tid = threadIdx.x;
  const int wave = tid >> 5, lane = tid & 31;
  const int lrow = lane & 15, lhalf = lane >> 4;
  const int colBase = blockIdx.y * 16;
  const int rowBase = blockIdx.x * 64 + wave * 16;
  const float* fcr = FC + (size_t)(rowBase + lrow) * Dd;
  const float* fpr = FP + (size_t)(rowBase + lrow) * Dd;
  const float* xr  = X  + (size_t)(rowBase + lrow) * Dd;

  v8f accO = {}, accG = {};

  // ---- segment 1: feat_c vs agg_W[:, 0:300]
  stage_panel_cmajor(pB1, aggW, 900, 0, colBase, tid);
  __syncthreads();
  for (int kk = 0; kk < Dd / 4; ++kk) {
    v2f a, b;
#pragma unroll
    for (int r = 0; r < 2; ++r) {
      int k = kk * 4 + r + 2 * lhalf;
      a[r] = fcr[k];
      b[r] = pB1[lrow * Dd + k];
    }
    accO = wmma4(a, b, accO);
  }
  __syncthreads();

  // ---- segment 2: feat_p vs agg_W[:, 300:600]
  stage_panel_cmajor(pB1, aggW, 900, 300, colBase, tid);
  __syncthreads();
  for (int kk = 0; kk < Dd / 4; ++kk) {
    v2f a, b;
#pragma unroll
    for (int r = 0; r < 2; ++r) {
      int k = kk * 4 + r + 2 * lhalf;
      a[r] = fpr[k];
      b[r] = pB1[lrow * Dd + k];
    }
    accO = wmma4(a, b, accO);
  }
  __syncthreads();

  // ---- segment 3: feat_in vs agg_W[:, 600:900]  +  gate GEMM vs H_W
  stage_panel_cmajor(pB1, aggW, 900, 600, colBase, tid);
  stage_panel_cmajor(pB2, HW, Dd, 0, colBase, tid);
  __syncthreads();
  for (int kk = 0; kk < Dd / 4; ++kk) {
    v2f a, b1, b2;
#pragma unroll
    for (int r = 0; r < 2; ++r) {
      int k = kk * 4 + r + 2 * lhalf;
      a[r] = xr[k];
      b1[r] = pB1[lrow * Dd + k];
      b2[r] = pB2[lrow * Dd + k];
    }
    accO = wmma4(a, b1, accO);
    accG = wmma4(a, b2, accG);
  }

  // ---- epilogue: rank-1 root term + bias, ELU, sigmoid gate mix
#pragma unroll
  for (int r = 0; r < 8; ++r) {
    int mm = rowBase + r + 8 * lhalf;
    int cc = colBase + lrow;
    if (cc < Dd) {
      float pre = accO[r] + ro[mm] * redot[cc] + aggb[cc];
      float act = (pre > 0.0f) ? pre : (expf(pre) - 1.0f);  // elu
      float g = 1.0f / (1.0f + expf(-(accG[r] + Hb[cc])));  // sigmoid
      float fi = X[(size_t)mm * Dd + cc];
      out[(size_t)mm * Dd + cc] = g * act + (1.0f - g) * fi;
    }
  }
}

// ---------------------------------------------------------------------------
// Workspace layout (floats): ~422 MB total
// ---------------------------------------------------------------------------
static const size_t SZ_H   = (size_t)BN * Dd;       // 33,792,000
static const size_t SZ_NN  = (size_t)BN * Nn;       // 12,390,400
static const size_t OFF_H    = 0;                   // h, reused as feat_c
static const size_t OFF_ASRC = OFF_H + SZ_H;
static const size_t OFF_ADST = OFF_ASRC + BN;
static const size_t OFF_RPRB = OFF_ADST + BN;
static const size_t OFF_R    = OFF_RPRB + BN;
static const size_t OFF_DV   = OFF_R + BN;
static const size_t OFF_RO   = OFF_DV + BN;
static const size_t OFF_RD   = OFF_RO + BN;         // re_dot (512 pad)
static const size_t OFF_A    = OFF_RD + 512;
static const size_t OFF_LL   = OFF_A + SZ_NN;       // becomes LL^{-1} in place
static const size_t OFF_AO   = OFF_LL + SZ_NN;
static const size_t OFF_FP   = OFF_AO + SZ_NN;

extern "C" void kernel_launch(void* const* d_in, const int* in_sizes, int n_in,
                              void* d_out, int out_size, void* d_ws,
                              size_t ws_size, hipStream_t stream) {
  const float* feat  = (const float*)d_in[0];
  const float* adj   = (const float*)d_in[1];
  const float* W     = (const float*)d_in[2];
  const float* wsrc  = (const float*)d_in[3];
  const float* wdst  = (const float*)d_in[4];
  const float* rootW = (const float*)d_in[5];
  const float* rootb = (const float*)d_in[6];
  const float* aggW  = (const float*)d_in[7];
  const float* aggb  = (const float*)d_in[8];
  const float* HW    = (const float*)d_in[9];
  const float* Hb    = (const float*)d_in[10];
  const float* remb  = (const float*)d_in[11];
  float* out = (float*)d_out;
  float* ws = (float*)d_ws;

  float* h     = ws + OFF_H;
  float* asrc  = ws + OFF_ASRC;
  float* adst  = ws + OFF_ADST;
  float* rprb  = ws + OFF_RPRB;
  float* Rw    = ws + OFF_R;
  float* dvw   = ws + OFF_DV;
  float* row   = ws + OFF_RO;
  float* redot = ws + OFF_RD;
  float* Aw    = ws + OFF_A;
  float* LLw   = ws + OFF_LL;
  float* AOw   = ws + OFF_AO;
  float* FPw   = ws + OFF_FP;

  // 1. h = X @ W (WMMA, async W panel)
  k_gemm_h<<<dim3(BN / 64, 19), 128, 0, stream>>>(feat, W, h);
  // 1b. root-emb rank-1 precompute
  k_redot<<<dim3(3), 128, 0, stream>>>(aggW, remb, redot);
  // 2. per-node reductions
  k_node<<<dim3(BN / 4), 128, 0, stream>>>(h, rootW, rootb, wsrc, wdst, asrc,
                                           adst, rprb);
  // 3. A, R, LL
  k_buildA<<<dim3(Bb), 128, 0, stream>>>(adj, asrc, adst, rprb, Aw, LLw, Rw);
  // 4. batched GJ inverse + root_out (writes d_out tail)
  k_invert<<<dim3(Bb), 256, 0, stream>>>(LLw, dvw, Rw, adj, row, out + SZ_H);
  // 5. attn_out
  k_attnout<<<dim3(Bb, Nn), 128, 0, stream>>>(Aw, LLw, dvw, adj, AOw);
  // 6. feat_c / feat_p dual WMMA GEMM (feat_c reuses h buffer)
  k_feat_cp<<<dim3((7 * 19 + 3) / 4, Bb), 128, 0, stream>>>(AOw, feat, h, FPw);
  // 7. fused output + gate WMMA GEMM with async-staged B panels
  k_final<<<dim3(BN / 64, 19), 128, 0, stream>>>(h, FPw, feat, aggW, aggb, HW,
                                                 Hb, row, redot, out);
}